// LightSourceConsistencyLoss_87986700026271
// MI455X (gfx1250) — compile-verified
//
#include <hip/hip_runtime.h>
#include <hip/hip_bf16.h>

// ---------------- problem constants ----------------
#define BN   16
#define CN   3
#define HN   512
#define WN   512
#define HWN  (HN * WN)          // 262144 = 2^18
#define NPIX (BN * HWN)         // 4194304 pixels
#define CHWN (CN * HWN)
#define RAD  8
#define KW   17                 // 2*RAD+1
#define NPART 4096

__constant__ const float kInvK2  = 1.0f / 289.0f;   // 1/(17*17)
__constant__ const float kEps    = 1e-5f;
__constant__ const float kThresh = 0.8f;

typedef float v2f __attribute__((ext_vector_type(2)));
typedef float v8f __attribute__((ext_vector_type(8)));

// f32 WMMA 16x16x4: D = A(16x4) * B(4x16) + C, full fp32 precision.
static __device__ __forceinline__ v8f wmma_f32_16x16x4(v2f a, v2f b, v8f c) {
    return __builtin_amdgcn_wmma_f32_16x16x4_f32(
        /*neg_a=*/false, a, /*neg_b=*/false, b,
        /*c_mod=*/(short)0, c, /*reuse_a=*/false, /*reuse_b=*/false);
}

// Per-tile context for the vertical banded-ones matmul. The 17-tap
// zero-padded vertical window sum of a 16x16 tile is
//   out[m][n] = sum_{K: K-m in [0,16]} in[y0-8+K][x0+n]
// done as 8 chained K=4 f32 WMMAs (K = 0..31).
// Boundary handling is folded into the A matrix: load addresses are CLAMPED
// (always in-bounds, unconditional loads -> no EXEC branching) and the band
// coefficient is zeroed when the source row is outside [0,H).
struct VTileCtx {
    int off[16];   // plane-relative element offsets, 2 per K-block
    v2f af[8];     // A fragments (band * in-range predicate)
};

static __device__ __forceinline__ void vtile_init(VTileCtx& c, int y0, int x0, int lane) {
    const int m     = lane & 15;
    const int khalf = (lane < 16) ? 0 : 2;   // A/B VGPR K split per ISA layout
    const int col   = x0 + m;
#pragma unroll
    for (int kb = 0; kb < 8; ++kb) {
        const int K0 = kb * 4 + khalf;
        const int K1 = K0 + 1;
        const int r0 = y0 - RAD + K0;
        const int r1 = y0 - RAD + K1;
        const bool in0 = (r0 >= 0) && (r0 < HN);
        const bool in1 = (r1 >= 0) && (r1 < HN);
        const int rc0 = min(max(r0, 0), HN - 1);
        const int rc1 = min(max(r1, 0), HN - 1);
        c.off[2 * kb + 0] = rc0 * WN + col;
        c.off[2 * kb + 1] = rc1 * WN + col;
        c.af[kb][0] = (in0 && K0 >= m && K0 <= m + 16) ? 1.0f : 0.0f;
        c.af[kb][1] = (in1 && K1 >= m && K1 <= m + 16) ? 1.0f : 0.0f;
    }
}

static __device__ __forceinline__ v8f vtile_sum(const float* __restrict__ plane,
                                                const VTileCtx& c) {
    v8f acc = {0.f, 0.f, 0.f, 0.f, 0.f, 0.f, 0.f, 0.f};
#pragma unroll
    for (int kb = 0; kb < 8; ++kb) {
        v2f bf;
        bf[0] = plane[c.off[2 * kb + 0]];
        bf[1] = plane[c.off[2 * kb + 1]];
        acc = wmma_f32_16x16x4(c.af[kb], bf, acc);
    }
    return acc;
}

// ---------------- kernels ----------------

__global__ void k_init(int* __restrict__ flags, float* __restrict__ partials) {
    int i = blockIdx.x * blockDim.x + threadIdx.x;
    if (i < 2) flags[i] = 0;
    if (i < NPART) partials[i] = 0.0f;
}

// flags[0] = any(input<0), flags[1] = any(output<0)
__global__ void k_flags(const float* __restrict__ outp, const float* __restrict__ inp,
                        int* __restrict__ flags) {
    const int stride = gridDim.x * blockDim.x;
    bool negI = false, negO = false;
    for (int i = blockIdx.x * blockDim.x + threadIdx.x; i < BN * CHWN; i += stride) {
        negI |= (inp[i]  < 0.0f);
        negO |= (outp[i] < 0.0f);
    }
    if (negI) atomicOr(&flags[0], 1);
    if (negO) atomicOr(&flags[1], 1);
}

// guide = channel-mean of img, mask = (channel-max > 0.8)
__global__ void k_prep(const float* __restrict__ inp, const int* __restrict__ flags,
                       float* __restrict__ guide, float* __restrict__ mask) {
    int idx = blockIdx.x * blockDim.x + threadIdx.x;
    int b = idx >> 18;               // / HWN
    int p = idx & (HWN - 1);
    const float* src = inp + b * CHWN + p;
    float v0 = src[0], v1 = src[HWN], v2 = src[2 * HWN];
    if (flags[0]) { v0 = (v0 + 1.0f) * 0.5f; v1 = (v1 + 1.0f) * 0.5f; v2 = (v2 + 1.0f) * 0.5f; }
    float mx = fmaxf(v0, fmaxf(v1, v2));
    guide[idx] = (v0 + v1 + v2) * (1.0f / 3.0f);
    mask[idx]  = (mx > kThresh) ? 1.0f : 0.0f;
}

// Horizontal 17-tap zero-padded window sums of {g, m, g*m, g*g} via LDS tile.
__global__ void k_hpass4(const float* __restrict__ g, const float* __restrict__ msk,
                         float* __restrict__ o0, float* __restrict__ o1,
                         float* __restrict__ o2, float* __restrict__ o3) {
    __shared__ float sg[256 + 2 * RAD];
    __shared__ float sm[256 + 2 * RAD];
    const int rowId = blockIdx.x >> 1;
    const int x0    = (blockIdx.x & 1) * 256;
    const int base  = rowId * WN;
    const int tid   = threadIdx.x;
    int x = x0 - RAD + tid;
    sg[tid] = (x >= 0 && x < WN) ? g[base + x]   : 0.0f;
    sm[tid] = (x >= 0 && x < WN) ? msk[base + x] : 0.0f;
    if (tid < 2 * RAD) {
        int x2 = x + 256, i2 = tid + 256;
        sg[i2] = (x2 >= 0 && x2 < WN) ? g[base + x2]   : 0.0f;
        sm[i2] = (x2 >= 0 && x2 < WN) ? msk[base + x2] : 0.0f;
    }
    __syncthreads();
    float a0 = 0.f, a1 = 0.f, a2 = 0.f, a3 = 0.f;
#pragma unroll
    for (int d = 0; d < KW; ++d) {
        float gv = sg[tid + d], mv = sm[tid + d];
        a0 += gv; a1 += mv; a2 += gv * mv; a3 += gv * gv;
    }
    const int off = base + x0 + tid;
    o0[off] = a0; o1[off] = a1; o2[off] = a2; o3[off] = a3;
}

// Horizontal window sums of {a, b}.
__global__ void k_hpass2(const float* __restrict__ a, const float* __restrict__ b,
                         float* __restrict__ o0, float* __restrict__ o1) {
    __shared__ float sa[256 + 2 * RAD];
    __shared__ float sb[256 + 2 * RAD];
    const int rowId = blockIdx.x >> 1;
    const int x0    = (blockIdx.x & 1) * 256;
    const int base  = rowId * WN;
    const int tid   = threadIdx.x;
    int x = x0 - RAD + tid;
    sa[tid] = (x >= 0 && x < WN) ? a[base + x] : 0.0f;
    sb[tid] = (x >= 0 && x < WN) ? b[base + x] : 0.0f;
    if (tid < 2 * RAD) {
        int x2 = x + 256, i2 = tid + 256;
        sa[i2] = (x2 >= 0 && x2 < WN) ? a[base + x2] : 0.0f;
        sb[i2] = (x2 >= 0 && x2 < WN) ? b[base + x2] : 0.0f;
    }
    __syncthreads();
    float s0 = 0.f, s1 = 0.f;
#pragma unroll
    for (int d = 0; d < KW; ++d) { s0 += sa[tid + d]; s1 += sb[tid + d]; }
    const int off = base + x0 + tid;
    o0[off] = s0; o1[off] = s1;
}

// WMMA vertical pass #1 + fused a/b. One 16x16 tile per wave (4 waves/block).
__global__ void k_vab(const float* __restrict__ hI, const float* __restrict__ hP,
                      const float* __restrict__ hIp, const float* __restrict__ hII,
                      float* __restrict__ oA, float* __restrict__ oB) {
    const int wave = threadIdx.x >> 5;
    const int lane = threadIdx.x & 31;
    const int t    = blockIdx.x * 4 + wave;       // 16384 tiles = 16 * 32 * 32
    const int bI   = t >> 10;
    const int rem  = t & 1023;
    const int y0   = (rem >> 5) << 4;
    const int x0   = (rem & 31) << 4;

    VTileCtx ctx;
    vtile_init(ctx, y0, x0, lane);
    const int pb = bI * HWN;

    v8f sI  = vtile_sum(hI  + pb, ctx);
    v8f sP  = vtile_sum(hP  + pb, ctx);
    v8f sIp = vtile_sum(hIp + pb, ctx);
    v8f sII = vtile_sum(hII + pb, ctx);

    const int m      = lane & 15;
    const int rowAdd = (lane < 16) ? 0 : 8;       // C/D layout: VGPR r -> rows r / r+8
    const int col    = x0 + m;
#pragma unroll
    for (int r = 0; r < 8; ++r) {
        float mI  = sI[r]  * kInvK2;
        float mP  = sP[r]  * kInvK2;
        float mIp = sIp[r] * kInvK2;
        float mII = sII[r] * kInvK2;
        float av  = (mIp - mI * mP) / (mII - mI * mI + kEps);
        float bv  = mP - av * mI;
        const int off = pb + (y0 + r + rowAdd) * WN + col;
        oA[off] = av;
        oB[off] = bv;
    }
}

// WMMA vertical pass #2 + fused q = mean_a*guide + mean_b, clamped.
__global__ void k_vfinal(const float* __restrict__ hA, const float* __restrict__ hB,
                         const float* __restrict__ guide, float* __restrict__ mask) {
    const int wave = threadIdx.x >> 5;
    const int lane = threadIdx.x & 31;
    const int t    = blockIdx.x * 4 + wave;
    const int bI   = t >> 10;
    const int rem  = t & 1023;
    const int y0   = (rem >> 5) << 4;
    const int x0   = (rem & 31) << 4;

    VTileCtx ctx;
    vtile_init(ctx, y0, x0, lane);
    const int pb = bI * HWN;

    v8f sA = vtile_sum(hA + pb, ctx);
    v8f sB = vtile_sum(hB + pb, ctx);

    const int m      = lane & 15;
    const int rowAdd = (lane < 16) ? 0 : 8;
    const int col    = x0 + m;
#pragma unroll
    for (int r = 0; r < 8; ++r) {
        const int off = pb + (y0 + r + rowAdd) * WN + col;
        float q = sA[r] * kInvK2 * guide[off] + sB[r] * kInvK2;
        mask[off] = fminf(fmaxf(q, 0.0f), 1.0f);
    }
}

// Masked-L1 partial sums: sum over pixels,channels of mask * |out01 - img|.
__global__ void k_loss(const float* __restrict__ outp, const float* __restrict__ inp,
                       const float* __restrict__ mask, const int* __restrict__ flags,
                       float* __restrict__ partials) {
    __shared__ float red[256];
    const int tid = threadIdx.x;
    const int stride = gridDim.x * blockDim.x;
    const int fi = flags[0], fo = flags[1];
    float acc = 0.0f;
    for (int idx = blockIdx.x * blockDim.x + tid; idx < NPIX; idx += stride) {
        const int b = idx >> 18;
        const int p = idx & (HWN - 1);
        const float mv = mask[idx];
        const int base = b * CHWN + p;
        // stream-ahead prefetch (emits global_prefetch_b8; speculative, OOB dropped)
        __builtin_prefetch(&inp[base + stride], 0, 0);
        __builtin_prefetch(&outp[base + stride], 0, 0);
#pragma unroll
        for (int c = 0; c < CN; ++c) {
            float iv = inp[base + c * HWN];
            float ov = outp[base + c * HWN];
            if (fi) iv = (iv + 1.0f) * 0.5f;
            if (fo) ov = (ov + 1.0f) * 0.5f;
            acc += mv * fabsf(ov - iv);
        }
    }
    red[tid] = acc;
    __syncthreads();
    for (int s = 128; s > 0; s >>= 1) {
        if (tid < s) red[tid] += red[tid + s];
        __syncthreads();
    }
    if (tid == 0) partials[blockIdx.x] = red[0];
}

__global__ void k_final(const float* __restrict__ partials, float* __restrict__ out) {
    __shared__ float red[256];
    const int tid = threadIdx.x;
    float acc = 0.0f;
    for (int i = tid; i < NPART; i += 256) acc += partials[i];
    red[tid] = acc;
    __syncthreads();
    for (int s = 128; s > 0; s >>= 1) {
        if (tid < s) red[tid] += red[tid + s];
        __syncthreads();
    }
    if (tid == 0) out[0] = red[0] * (1.0f / (float)(BN * CN * HN * WN));
}

// ---------------- launcher ----------------
extern "C" void kernel_launch(void* const* d_in, const int* in_sizes, int n_in,
                              void* d_out, int out_size, void* d_ws, size_t ws_size,
                              hipStream_t stream) {
    const float* outp = (const float*)d_in[0];   // "output" (setup_inputs order)
    const float* inp  = (const float*)d_in[1];   // "input"
    float* res = (float*)d_out;

    float* wsf    = (float*)d_ws;
    float* gGuide = wsf + 0 * (size_t)NPIX;
    float* gMask  = wsf + 1 * (size_t)NPIX;      // initial mask, then final mask
    float* t0     = wsf + 2 * (size_t)NPIX;
    float* t1     = wsf + 3 * (size_t)NPIX;
    float* t2     = wsf + 4 * (size_t)NPIX;
    float* t3     = wsf + 5 * (size_t)NPIX;
    float* t4     = wsf + 6 * (size_t)NPIX;
    float* t5     = wsf + 7 * (size_t)NPIX;
    int*   flags    = (int*)(wsf + 8 * (size_t)NPIX);
    float* partials = wsf + 8 * (size_t)NPIX + 8;

    k_init  <<<(NPART + 255) / 256, 256, 0, stream>>>(flags, partials);
    k_flags <<<4096, 256, 0, stream>>>(outp, inp, flags);
    k_prep  <<<NPIX / 256, 256, 0, stream>>>(inp, flags, gGuide, gMask);
    k_hpass4<<<BN * HN * 2, 256, 0, stream>>>(gGuide, gMask, t0, t1, t2, t3);
    k_vab   <<<4096, 128, 0, stream>>>(t0, t1, t2, t3, t4, t5);
    k_hpass2<<<BN * HN * 2, 256, 0, stream>>>(t4, t5, t0, t1);
    k_vfinal<<<4096, 128, 0, stream>>>(t0, t1, gGuide, gMask);
    k_loss  <<<NPART, 256, 0, stream>>>(outp, inp, gMask, flags, partials);
    k_final <<<1, 256, 0, stream>>>(partials, res);
}